// ECLoss_16947940950787
// MI455X (gfx1250) — compile-verified
//
#include <hip/hip_runtime.h>
#include <math.h>

typedef __attribute__((ext_vector_type(2))) float v2f;
typedef __attribute__((ext_vector_type(8))) float v8f;

#define NUM_BINS 15
#define NQ   10000
#define NTR  20000
#define NTE  10000
#define NB   1024
#define BW   0.3f
#define INV2BW2 (1.0f / (2.0f * BW * BW))                   /* 5.5555553f  */
#define LOG2E 1.44269504088896340736f
#define KL2 (INV2BW2 * LOG2E)                               /* 8.0149727f  */
#define KDE_NORM (2.0f * 3.14159265358979323846f * BW * BW) /* 0.56548667f */
#define VOLUME 100.0f

/* workspace layout (floats) */
#define WS_CNT 0
#define WS_TR  1
#define WS_TE  16
#define WS_INT 32
#define WS_CW  64

/* ---------------- phase 0: batch correctness weights + zero integrals ------ */
__global__ void ec_setup(const int* __restrict__ y, const int* __restrict__ pred,
                         float* __restrict__ ws) {
    __shared__ float sred[1024];
    int i = threadIdx.x;
    float c = (y[i] == pred[i]) ? 1.0f : 0.0f;
    ws[WS_CW + i] = c;
    sred[i] = c;
    __syncthreads();
    for (int s = 512; s > 0; s >>= 1) {
        if (i < s) sred[i] += sred[i + s];
        __syncthreads();
    }
    if (i == 0) ws[WS_CNT] = sred[0];
    if (i < NUM_BINS) ws[WS_INT + i] = 0.0f;
}

/* ---------------- phase 0: per-bin histogram rates ------------------------- */
__global__ void ec_rates(const float* __restrict__ trainp,
                         const float* __restrict__ testp,
                         float* __restrict__ ws) {
    __shared__ int sred[256];
    int bin = blockIdx.x;
    const float* p = (blockIdx.y == 0) ? trainp : testp;
    int n = (blockIdx.y == 0) ? NTR : NTE;
    float lo = (float)bin / 15.0f;
    float hi = (float)(bin + 1) / 15.0f;
    int c = 0;
    for (int i = threadIdx.x; i < n; i += 256) {
        float v = p[i];
        c += (v > lo && v <= hi) ? 1 : 0;
    }
    sred[threadIdx.x] = c;
    __syncthreads();
    for (int s = 128; s > 0; s >>= 1) {
        if (threadIdx.x < s) sred[threadIdx.x] += sred[threadIdx.x + s];
        __syncthreads();
    }
    if (threadIdx.x == 0)
        ws[1 + 15 * blockIdx.y + bin] = (float)sred[0] / (float)n;
}

/* ---------------- main pairwise KDE loop (WMMA f32 16x16x4, full K=4) ------
 * The complete exp2 argument is produced by the WMMA itself:
 *   arg(M,N) = (2K*qx, 2K*qy, -K*|q|^2, 1) . (px, py, 1, -K*|p|^2)
 *            = 2^...  with K = INV2BW2*log2(e)
 * so the epilogue is exp + accumulate only.                                 */
__device__ __forceinline__ void kde_loop(const float2* __restrict__ data, int nchunks,
                                         int col, int half, v2f A,
                                         const float* __restrict__ cw, v8f& acc) {
    for (int c = 0; c < nchunks; ++c) {
        float2 p = data[c * 16 + col];
        float pn = fmaf(p.x, p.x, p.y * p.y);
        v2f B;
        B.x = half ? 1.0f : p.x;            /* K=0 rows | K=2 row of ones   */
        B.y = half ? (-KL2 * pn) : p.y;     /* K=1 rows | K=3 = -K*|p|^2    */
        v8f z = {0.f, 0.f, 0.f, 0.f, 0.f, 0.f, 0.f, 0.f};
        v8f d = __builtin_amdgcn_wmma_f32_16x16x4_f32(
            false, A, false, B, (short)0, z, false, false);
        float w = 1.0f;
        if (cw) w = cw[c * 16 + col];
#pragma unroll
        for (int r = 0; r < 8; ++r) {
            float e = __builtin_amdgcn_exp2f(d[r]);
            acc[r] = cw ? fmaf(e, w, acc[r]) : (acc[r] + e);
        }
    }
}

__global__ void ec_main(const float2* __restrict__ mc,
                        const float2* __restrict__ train,
                        const float2* __restrict__ test,
                        const float2* __restrict__ batch,
                        const float* __restrict__ W,
                        const float* __restrict__ bvec,
                        const float* __restrict__ ws,
                        float* __restrict__ integral) {
    const int wave = threadIdx.x >> 5;
    const int lane = threadIdx.x & 31;
    const int tile = blockIdx.x * 8 + wave;
    const int bin  = blockIdx.y;
    if (tile >= NQ / 16) return;              /* wave-uniform; no barriers used */

    const int half  = lane >> 4;
    const int col   = lane & 15;
    const int qbase = tile * 16;
    const float2* q = mc + (size_t)bin * NQ;

    /* query points kept only for the finalize softmax (lanes 0/16) */
    float2 qpt[8];
#pragma unroll
    for (int r = 0; r < 8; ++r) qpt[r] = q[qbase + r + 8 * half];

    /* A-matrix 16x4 (loop-invariant):
       lanes 0-15  (K=0,1): 2K*qx, 2K*qy
       lanes 16-31 (K=2,3): -K*|q|^2, 1                                    */
    float2 aq = q[qbase + col];
    float qn = fmaf(aq.x, aq.x, aq.y * aq.y);
    v2f A;
    A.x = half ? (-KL2 * qn) : (2.0f * KL2 * aq.x);
    A.y = half ? 1.0f : (2.0f * KL2 * aq.y);

    v8f accTr = {0.f,0.f,0.f,0.f,0.f,0.f,0.f,0.f};
    v8f accTe = {0.f,0.f,0.f,0.f,0.f,0.f,0.f,0.f};
    v8f accW  = {0.f,0.f,0.f,0.f,0.f,0.f,0.f,0.f};

    kde_loop(train, NTR / 16, col, half, A, nullptr,    accTr);
    kde_loop(test,  NTE / 16, col, half, A, nullptr,    accTe);
    kde_loop(batch, NB  / 16, col, half, A, ws + WS_CW, accW);

    /* reduce over the 16 columns (within each half-wave) */
#pragma unroll
    for (int r = 0; r < 8; ++r) {
        for (int m = 1; m < 16; m <<= 1) {
            accTr[r] += __shfl_xor(accTr[r], m, 32);
            accTe[r] += __shfl_xor(accTe[r], m, 32);
            accW[r]  += __shfl_xor(accW[r],  m, 32);
        }
    }

    const float cnt     = ws[WS_CNT];
    const float p_y     = cnt * (1.0f / (float)NB);
    const float tr_rate = ws[WS_TR + bin];
    const float te_rate = ws[WS_TE + bin];
    const float lo = (float)bin / 15.0f;
    const float hi = (float)(bin + 1) / 15.0f;
    const float sc_tr = 1.0f / ((float)NTR * KDE_NORM);
    const float sc_te = 1.0f / ((float)NTE * KDE_NORM);
    const float sc_w  = 1.0f / (fmaxf(cnt, 1.0f) * KDE_NORM);

    float partial = 0.0f;
    if (col == 0) {   /* lanes 0 (rows 0-7) and 16 (rows 8-15) finalize */
#pragma unroll
        for (int r = 0; r < 8; ++r) {
            float kde_tr = accTr[r] * sc_tr;
            float kde_te = accTe[r] * sc_te;
            float kde_w  = accW[r]  * sc_w;
            float p_hs = kde_w * p_y / (kde_tr + 1e-8f);
            p_hs = fminf(fmaxf(p_hs, 0.0f), 1.0f);

            /* hat_s = max softmax(x@W + b) = 1 / sum(exp(l - max)) */
            float qx = qpt[r].x, qy = qpt[r].y;
            float lg[10];
            float mx = -3.0e38f;
#pragma unroll
            for (int cc = 0; cc < 10; ++cc) {
                float l = fmaf(qx, W[cc], fmaf(qy, W[10 + cc], bvec[cc]));
                lg[cc] = l;
                mx = fmaxf(mx, l);
            }
            float se = 0.0f;
#pragma unroll
            for (int cc = 0; cc < 10; ++cc)
                se += __builtin_amdgcn_exp2f((lg[cc] - mx) * LOG2E);
            float hat = 1.0f / se;
            float idx = (hat >= lo && hat <= hi) ? 1.0f : 0.0f;

            float d_t = (te_rate > 0.0f) ? kde_te / te_rate : 0.0f;
            float d_s = (tr_rate > 0.0f) ? kde_tr / tr_rate : 0.0f;
            partial += p_hs * idx * (d_t - d_s);
        }
    }
    partial += __shfl_xor(partial, 16, 32);
    if (lane == 0) atomicAdd(&integral[bin], partial);
}

/* ---------------- final scalar: ec = sum_b te>0 ? te*|mean*VOL| : 0 -------- */
__global__ void ec_final(const float* __restrict__ ws, float* __restrict__ out) {
    if (threadIdx.x == 0 && blockIdx.x == 0) {
        float ec = 0.0f;
        for (int b = 0; b < NUM_BINS; ++b) {
            float te = ws[WS_TE + b];
            float integ = ws[WS_INT + b] * (VOLUME / (float)NQ);
            if (te > 0.0f) ec += te * fabsf(integ);
        }
        *out = ec;
    }
}

extern "C" void kernel_launch(void* const* d_in, const int* in_sizes, int n_in,
                              void* d_out, int out_size, void* d_ws, size_t ws_size,
                              hipStream_t stream) {
    (void)in_sizes; (void)n_in; (void)out_size; (void)ws_size;
    const float2* batch_x = (const float2*)d_in[0];
    const int*    y       = (const int*)d_in[1];
    const int*    pred    = (const int*)d_in[2];
    const float*  train_p = (const float*)d_in[3];
    const float*  test_p  = (const float*)d_in[4];
    const float2* train_x = (const float2*)d_in[5];
    const float2* test_x  = (const float2*)d_in[6];
    const float*  W       = (const float*)d_in[7];
    const float*  bv      = (const float*)d_in[8];
    const float2* mc      = (const float2*)d_in[9];
    float* ws  = (float*)d_ws;
    float* out = (float*)d_out;

    ec_setup<<<1, 1024, 0, stream>>>(y, pred, ws);
    ec_rates<<<dim3(NUM_BINS, 2), 256, 0, stream>>>(train_p, test_p, ws);
    ec_main<<<dim3((NQ / 16 + 7) / 8, NUM_BINS), 256, 0, stream>>>(
        mc, train_x, test_x, batch_x, W, bv, ws, ws + WS_INT);
    ec_final<<<1, 32, 0, stream>>>(ws, out);
}